// Discriminator_9294309228627
// MI455X (gfx1250) — compile-verified
//
#include <hip/hip_runtime.h>
#include <hip/hip_bf16.h>

#define HDIM   1024
#define N_IDX  65536
#define N_SEG  8192
#define KCHUNK 512
#define LSTRIDE (KCHUNK + 4)   // 516: %64==4 -> 4-bank skew per row, 16B aligned

typedef float v2f __attribute__((ext_vector_type(2)));
typedef float v8f __attribute__((ext_vector_type(8)));
typedef int   v4i __attribute__((ext_vector_type(4)));

#if __has_builtin(__builtin_amdgcn_global_load_async_to_lds_b128)
#define ASYNC_LDS_OK 1
#endif

// 16B global -> LDS copy: async (no VGPR round-trip, ASYNCcnt) when available.
__device__ __forceinline__ void cp16_g2lds(float* lds, const float* gptr) {
#ifdef ASYNC_LDS_OK
    __builtin_amdgcn_global_load_async_to_lds_b128(
        (__attribute__((address_space(1))) v4i*)gptr,
        (__attribute__((address_space(3))) v4i*)lds,
        /*imm offset*/ 0, /*cpol*/ 0);
#else
    *(float4*)lds = *(const float4*)gptr;
#endif
}

__device__ __forceinline__ void wait_async_copies() {
#ifdef ASYNC_LDS_OK
#if __has_builtin(__builtin_amdgcn_s_wait_asynccnt)
    __builtin_amdgcn_s_wait_asynccnt(0);
#else
    asm volatile("s_wait_asynccnt 0" ::: "memory");
#endif
#endif
}

// ---------------------------------------------------------------------------
// Kernel 1: segment boundaries via binary search over sorted segment_ids.
// starts[g] = lower_bound(segment_ids, g), g in [0, N_SEG]; starts[N_SEG]=N_IDX.
// ---------------------------------------------------------------------------
__global__ void seg_bounds_kernel(const int* __restrict__ segIds,
                                  int* __restrict__ starts) {
    int g = blockIdx.x * blockDim.x + threadIdx.x;
    if (g > N_SEG) return;
    int lo = 0, hi = N_IDX;
    while (lo < hi) {
        int mid = (lo + hi) >> 1;
        if (segIds[mid] < g) lo = mid + 1; else hi = mid;
    }
    starts[g] = lo;
}

// ---------------------------------------------------------------------------
// Kernel 2: one block per segment; gather rows of hid via indices and mean.
// 256 threads x float4 = 1024 columns. Coalesced 16B/lane row reads.
// ---------------------------------------------------------------------------
__global__ __launch_bounds__(256) void seg_mean_kernel(
    const float* __restrict__ hid, const int* __restrict__ idx,
    const int* __restrict__ starts, float* __restrict__ h) {
    const int g = blockIdx.x;
    const int s = starts[g];
    const int e = starts[g + 1];
    const int c = threadIdx.x * 4;
    float4 acc = make_float4(0.f, 0.f, 0.f, 0.f);
    for (int r = s; r < e; ++r) {
        const float4 v = *(const float4*)(hid + (size_t)idx[r] * HDIM + c);
        acc.x += v.x; acc.y += v.y; acc.z += v.z; acc.w += v.w;
    }
    const float inv = 1.0f / (float)((e - s) > 1 ? (e - s) : 1);
    acc.x *= inv; acc.y *= inv; acc.z *= inv; acc.w *= inv;
    *(float4*)(h + (size_t)g * HDIM + c) = acc;
}

// ---------------------------------------------------------------------------
// Kernel 3: h @ W_dense^T + b_dense -> exact-erf GELU -> * W_proj, partial
// N-reduction. Block = 128 threads (4 waves). Block tile: 16(M) x 64(N).
// A panel double-buffered in LDS via async global->LDS copies (ASYNCcnt),
// overlapping the chunk-1 copy with the chunk-0 WMMA loop.
// WMMA: V_WMMA_F32_16X16X4_F32, f32 native matrix path.
// ---------------------------------------------------------------------------
__global__ __launch_bounds__(128) void gemm_gelu_proj_kernel(
    const float* __restrict__ h,   const float* __restrict__ Wd,
    const float* __restrict__ bd,  const float* __restrict__ Wp,
    float* __restrict__ partial /* [N_SEG][64] */) {
    __shared__ float aTile[2][16 * LSTRIDE];   // 2 x 33,024 B

    const int tid   = threadIdx.x;
    const int wave  = tid >> 5;
    const int lane  = tid & 31;
    const int g0    = blockIdx.x * 16;       // M tile base (segment rows)
    const int nBase = blockIdx.y * 64 + wave * 16;
    const int nLane = lane & 15;
    const int klo   = (lane >> 4) << 1;      // A/B per-lane K sub-offset {0,2}

    const float* wRow = Wd + (size_t)(nBase + nLane) * HDIM;   // W_dense[n][k]

    // issue cooperative async copy of a 16 x 512 A panel into aTile[buf]
    auto load_panel = [&](int kc, int buf) {
        for (int i = tid * 4; i < 16 * KCHUNK; i += 128 * 4) {
            const int r = i >> 9;            // /512
            const int c = i & (KCHUNK - 1);
            cp16_g2lds(&aTile[buf][r * LSTRIDE + c],
                       h + (size_t)(g0 + r) * HDIM + kc + c);
        }
    };

    v8f acc = {};
    auto compute_chunk = [&](int kc, int buf) {
        const float* aRowB = aTile[buf] + nLane * LSTRIDE;  // A: m = lane&15
        const float* wChunk = wRow + kc;
#pragma unroll 8
        for (int k = 0; k < KCHUNK; k += 4) {
            v2f a = *(const v2f*)(aRowB + k + klo);     // ds_load_b64
            v2f b = *(const v2f*)(wChunk + k + klo);    // global_load_b64 (L2 hot)
            acc = __builtin_amdgcn_wmma_f32_16x16x4_f32(
                false, a, false, b, (short)0, acc, false, false);
        }
    };

    // ---- pipelined schedule (HDIM == 2*KCHUNK) ----
    load_panel(0, 0);
    __builtin_prefetch(wRow + KCHUNK, 0, 1);   // warm chunk-1 B lines
    wait_async_copies();
    __syncthreads();

    load_panel(KCHUNK, 1);     // async: overlaps with the 128-WMMA loop below
    compute_chunk(0, 0);

    wait_async_copies();
    __syncthreads();
    compute_chunk(KCHUNK, 1);

    // Epilogue: +b_dense, exact GELU, *W_proj, reduce over the 16 N lanes.
    const float bdv = bd[nBase + nLane];
    const float wpv = Wp[nBase + nLane];
    float p[8];
#pragma unroll
    for (int r = 0; r < 8; ++r) {
        const float x = acc[r] + bdv;
        const float zg = 0.5f * x * (1.0f + erff(x * 0.70710678118654752f));
        float s = zg * wpv;
        // C/D layout: lanes 0-15 -> M=r, lanes 16-31 -> M=r+8; reduce each half
        s += __shfl_xor(s, 1, 32);
        s += __shfl_xor(s, 2, 32);
        s += __shfl_xor(s, 4, 32);
        s += __shfl_xor(s, 8, 32);
        p[r] = s;
    }
    const int nChunk = nBase >> 4;           // 0..63
    if (nLane == 0) {                        // lanes 0 and 16
        const int mhi = (lane >> 4) << 3;    // 0 or 8
#pragma unroll
        for (int r = 0; r < 8; ++r)
            partial[(size_t)(g0 + r + mhi) * 64 + nChunk] = p[r];
    }
}

// ---------------------------------------------------------------------------
// Kernel 4: out[g] = b_proj + sum_{64} partial[g][*]
// ---------------------------------------------------------------------------
__global__ __launch_bounds__(256) void proj_reduce_kernel(
    const float* __restrict__ partial, const float* __restrict__ bp,
    float* __restrict__ out) {
    const int g = blockIdx.x * blockDim.x + threadIdx.x;
    if (g >= N_SEG) return;
    const float4* p = (const float4*)(partial + (size_t)g * 64);
    float s = 0.f;
#pragma unroll
    for (int i = 0; i < 16; ++i) {
        const float4 v = p[i];
        s += v.x + v.y + v.z + v.w;
    }
    out[g] = s + bp[0];
}

// ---------------------------------------------------------------------------
extern "C" void kernel_launch(void* const* d_in, const int* in_sizes, int n_in,
                              void* d_out, int out_size, void* d_ws, size_t ws_size,
                              hipStream_t stream) {
    const float* hid    = (const float*)d_in[0];  // [8*4096, 1024]
    const int*   idx    = (const int*)  d_in[1];  // [65536]
    const int*   segIds = (const int*)  d_in[2];  // [65536] sorted
    const float* Wd     = (const float*)d_in[3];  // [1024, 1024]
    const float* bd     = (const float*)d_in[4];  // [1024]
    const float* Wp     = (const float*)d_in[5];  // [1024]
    const float* bp     = (const float*)d_in[6];  // [1]
    float*       out    = (float*)d_out;          // [8192]

    // Workspace layout (aligned to 256B)
    char* ws = (char*)d_ws;
    int*   starts  = (int*)ws;                                  // 8193 ints
    float* h       = (float*)(ws + ((sizeof(int) * (N_SEG + 1) + 255) & ~(size_t)255));
    float* partial = h + (size_t)N_SEG * HDIM;                  // [8192][64]

    seg_bounds_kernel<<<(N_SEG + 1 + 255) / 256, 256, 0, stream>>>(segIds, starts);
    seg_mean_kernel<<<N_SEG, 256, 0, stream>>>(hid, idx, starts, h);
    gemm_gelu_proj_kernel<<<dim3(N_SEG / 16, HDIM / 64), 128, 0, stream>>>(
        h, Wd, bd, Wp, partial);
    proj_reduce_kernel<<<(N_SEG + 255) / 256, 256, 0, stream>>>(partial, bp, out);
}